// IAIDNNnet_21689584845494
// MI455X (gfx1250) — compile-verified
//
#include <hip/hip_runtime.h>
#include <hip/hip_bf16.h>

// Problem constants (from reference)
constexpr int cN = 512;   // batch
constexpr int cK = 8;     // users
constexpr int cR = 8;     // NR
constexpr int cT = 64;    // NT
constexpr int cD = 4;     // D
constexpr int cITERS = 5;
constexpr float PPOW = 100.0f;
constexpr float SIG  = 1e-7f;

typedef __attribute__((ext_vector_type(2))) float v2f;
typedef __attribute__((ext_vector_type(8))) float v8f;

// ---------------- complex helpers (float2 = {re, im}) ----------------
__device__ __forceinline__ float2 cadd(float2 a, float2 b) { return make_float2(a.x + b.x, a.y + b.y); }
__device__ __forceinline__ float2 csub(float2 a, float2 b) { return make_float2(a.x - b.x, a.y - b.y); }
__device__ __forceinline__ float2 cmul(float2 a, float2 b) {
    return make_float2(a.x * b.x - a.y * b.y, a.x * b.y + a.y * b.x);
}
// a * conj(b)
__device__ __forceinline__ float2 cmulc(float2 a, float2 b) {
    return make_float2(a.x * b.x + a.y * b.y, a.y * b.x - a.x * b.y);
}
__device__ __forceinline__ float2 cfma(float2 a, float2 b, float2 c) { return cadd(c, cmul(a, b)); }
__device__ __forceinline__ float2 cinv(float2 a) {
    float s = 1.0f / (a.x * a.x + a.y * a.y);
    return make_float2(a.x * s, -a.y * s);
}

// ---------------- CDNA5 fp32 WMMA: D(16x16) = A(16x4) * B(4x16) + C ----------------
__device__ __forceinline__ v8f wmma4(v2f a, v2f b, v8f c) {
    return __builtin_amdgcn_wmma_f32_16x16x4_f32(false, a, false, b, (short)0, c, false, false);
}

// =====================================================================
// Kernel 1: per-n small linear algebra chain.
// block = 64 threads, grid = N.
// Produces: P = (w_k * YWYh) @ H            [n][k][r][u]   (8x64 per (n,k))
//           G = Hd^H @ YW                   [n][k][t][d]   (64x4 per (n,k))
//           W1 = Yp @ G   (B for k3 WMMA)   [n][t][k*4+d]  (64x32 per n)
//           XpG = Xp @ G, ZpG = Zp @ G      [n][k][t][d]
//           beta_u[n]
// =====================================================================
__global__ __launch_bounds__(64) void k1_small(
    const float* __restrict__ Hre, const float* __restrict__ Him,
    const float* __restrict__ Vre, const float* __restrict__ Vim,
    const float* __restrict__ Ypre, const float* __restrict__ Ypim,
    const float* __restrict__ Xpre, const float* __restrict__ Xpim,
    const float* __restrict__ Zpre, const float* __restrict__ Zpim,
    const float* __restrict__ w,
    float2* __restrict__ P, float2* __restrict__ G,
    float2* __restrict__ W1, float2* __restrict__ XpG, float2* __restrict__ ZpG,
    float2* __restrict__ beta_u)
{
    const int n = blockIdx.x;
    const int tid = threadIdx.x;

    __shared__ float2 sV[cK * cT * cD];        // 2048 cplx
    __shared__ float2 sHV[cK * cK * cR * cD];  // 2048 cplx
    __shared__ float2 sG[cK * cT * cD];        // 2048 cplx
    __shared__ float2 sYW[cK * cR * cD];       // 256 cplx
    __shared__ float2 sWY[cK * cR * cR];       // 512 cplx (w-weighted YWYh)
    __shared__ float2 strace[cK];
    __shared__ float red[64];
    __shared__ float sbetaU;

    // ---- load V[n] into LDS + ||V||^2 reduction for beta_U ----
    const size_t vbase = (size_t)n * (cK * cT * cD);
    float nrm = 0.f;
    for (int idx = tid; idx < cK * cT * cD; idx += 64) {
        float2 v = make_float2(Vre[vbase + idx], Vim[vbase + idx]);
        sV[idx] = v;
        nrm += v.x * v.x + v.y * v.y;
    }
    red[tid] = nrm;
    __syncthreads();
    for (int s = 32; s > 0; s >>= 1) {
        if (tid < s) red[tid] += red[tid + s];
        __syncthreads();
    }
    if (tid == 0) sbetaU = (SIG / PPOW) * red[0];
    __syncthreads();

    // ---- HV[i][j] = H[n,i] (8x64) @ V[n,j] (64x4); thread = (i,j) pair ----
    {
        const int i = tid >> 3, j = tid & 7;
        float2 acc[cR][cD];
        #pragma unroll
        for (int r = 0; r < cR; ++r)
            #pragma unroll
            for (int d = 0; d < cD; ++d) acc[r][d] = make_float2(0.f, 0.f);
        const size_t hb = ((size_t)n * cK + i) * (cR * cT);
        for (int t = 0; t < cT; ++t) {
            float2 vv[cD];
            #pragma unroll
            for (int d = 0; d < cD; ++d) vv[d] = sV[(j * cT + t) * cD + d];
            #pragma unroll
            for (int r = 0; r < cR; ++r) {
                float2 h = make_float2(Hre[hb + r * cT + t], Him[hb + r * cT + t]);
                #pragma unroll
                for (int d = 0; d < cD; ++d) acc[r][d] = cfma(h, vv[d], acc[r][d]);
            }
        }
        #pragma unroll
        for (int r = 0; r < cR; ++r)
            #pragma unroll
            for (int d = 0; d < cD; ++d)
                sHV[((i * cK + j) * cR + r) * cD + d] = acc[r][d];
    }
    __syncthreads();

    // ---- per-k solve chain (threads 0..7) ----
    if (tid < cK) {
        const int i = tid;
        float2 U[cR][cR];
        #pragma unroll
        for (int r = 0; r < cR; ++r)
            #pragma unroll
            for (int s = 0; s < cR; ++s) U[r][s] = make_float2(0.f, 0.f);
        for (int j = 0; j < cK; ++j)
            for (int r = 0; r < cR; ++r)
                for (int s = 0; s < cR; ++s)
                    for (int d = 0; d < cD; ++d)
                        U[r][s] = cadd(U[r][s],
                            cmulc(sHV[((i * cK + j) * cR + r) * cD + d],
                                  sHV[((i * cK + j) * cR + s) * cD + d]));
        for (int r = 0; r < cR; ++r) U[r][r].x += sbetaU;

        float2 HV2[cR][cD], B[cR][cD];
        for (int r = 0; r < cR; ++r)
            for (int d = 0; d < cD; ++d)
                HV2[r][d] = B[r][d] = sHV[((i * cK + i) * cR + r) * cD + d];

        // Gauss-Jordan on Hermitian-PD U (no pivoting needed): B <- U^{-1} HV2 = Yv
        for (int p = 0; p < cR; ++p) {
            float2 ip = cinv(U[p][p]);
            for (int c = 0; c < cR; ++c) U[p][c] = cmul(U[p][c], ip);
            for (int c = 0; c < cD; ++c) B[p][c] = cmul(B[p][c], ip);
            for (int r = 0; r < cR; ++r) {
                if (r == p) continue;
                float2 f = U[r][p];
                for (int c = 0; c < cR; ++c) U[r][c] = csub(U[r][c], cmul(f, U[p][c]));
                for (int c = 0; c < cD; ++c) B[r][c] = csub(B[r][c], cmul(f, B[p][c]));
            }
        }

        // M4 = I4 - Yv^H @ HV2 ; Wm = inv(M4) via Gauss-Jordan
        float2 M4[cD][cD], Wm[cD][cD];
        for (int d = 0; d < cD; ++d)
            for (int e = 0; e < cD; ++e) {
                float2 s = make_float2(0.f, 0.f);
                for (int r = 0; r < cR; ++r) s = cadd(s, cmulc(HV2[r][e], B[r][d]));
                M4[d][e] = make_float2((d == e ? 1.f : 0.f) - s.x, -s.y);
                Wm[d][e] = make_float2(d == e ? 1.f : 0.f, 0.f);
            }
        for (int p = 0; p < cD; ++p) {
            float2 ip = cinv(M4[p][p]);
            for (int c = 0; c < cD; ++c) { M4[p][c] = cmul(M4[p][c], ip); Wm[p][c] = cmul(Wm[p][c], ip); }
            for (int r = 0; r < cD; ++r) {
                if (r == p) continue;
                float2 f = M4[r][p];
                for (int c = 0; c < cD; ++c) {
                    M4[r][c] = csub(M4[r][c], cmul(f, M4[p][c]));
                    Wm[r][c] = csub(Wm[r][c], cmul(f, Wm[p][c]));
                }
            }
        }

        // YW = Yv @ Wm ; YWYh = YW @ Yv^H (w-weighted) ; trace
        float2 YW[cR][cD];
        for (int r = 0; r < cR; ++r)
            for (int d = 0; d < cD; ++d) {
                float2 s = make_float2(0.f, 0.f);
                for (int e = 0; e < cD; ++e) s = cadd(s, cmul(B[r][e], Wm[e][d]));
                YW[r][d] = s;
                sYW[(i * cR + r) * cD + d] = s;
            }
        const float wk = w[i];
        float2 tr = make_float2(0.f, 0.f);
        for (int r = 0; r < cR; ++r)
            for (int s = 0; s < cR; ++s) {
                float2 acc = make_float2(0.f, 0.f);
                for (int d = 0; d < cD; ++d) acc = cadd(acc, cmulc(YW[r][d], B[s][d]));
                sWY[(i * cR + r) * cR + s] = make_float2(wk * acc.x, wk * acc.y);
                if (r == s) tr = cadd(tr, acc);
            }
        strace[i] = make_float2(wk * tr.x, wk * tr.y);
    }
    __syncthreads();

    if (tid == 0) {
        float2 s = make_float2(0.f, 0.f);
        for (int k = 0; k < cK; ++k) s = cadd(s, strace[k]);
        beta_u[n] = make_float2((SIG / PPOW) * s.x, (SIG / PPOW) * s.y);
    }

    // ---- P[k][r][u] = (wYWYh @ H)  and  G[k][t=u][d] = H^H @ YW ; thread = column u ----
    {
        const int u = tid;
        for (int k = 0; k < cK; ++k) {
            const size_t hb = ((size_t)n * cK + k) * (cR * cT);
            float2 hcol[cR];
            #pragma unroll
            for (int s = 0; s < cR; ++s)
                hcol[s] = make_float2(Hre[hb + s * cT + u], Him[hb + s * cT + u]);
            #pragma unroll
            for (int r = 0; r < cR; ++r) {
                float2 acc = make_float2(0.f, 0.f);
                #pragma unroll
                for (int s = 0; s < cR; ++s)
                    acc = cfma(sWY[(k * cR + r) * cR + s], hcol[s], acc);
                P[(((size_t)n * cK + k) * cR + r) * cT + u] = acc;
            }
            #pragma unroll
            for (int d = 0; d < cD; ++d) {
                float2 acc = make_float2(0.f, 0.f);
                #pragma unroll
                for (int r = 0; r < cR; ++r)
                    acc = cadd(acc, cmulc(sYW[(k * cR + r) * cD + d], hcol[r]));
                sG[(k * cT + u) * cD + d] = acc;
                G[(((size_t)n * cK + k) * cT + u) * cD + d] = acc;
            }
        }
    }
    __syncthreads();

    // ---- W1 = Yp @ G (layout [n][t][k*4+d]), XpG = Xp @ G, ZpG = Zp @ G ; thread = row t ----
    {
        const int t = tid;
        for (int k = 0; k < cK; ++k) {
            float2 aY[cD], aX[cD], aZ[cD];
            #pragma unroll
            for (int d = 0; d < cD; ++d) aY[d] = aX[d] = aZ[d] = make_float2(0.f, 0.f);
            const size_t rb = ((size_t)k * cT + t) * cT;
            for (int u = 0; u < cT; ++u) {
                float2 y = make_float2(Ypre[rb + u], Ypim[rb + u]);
                float2 x = make_float2(Xpre[rb + u], Xpim[rb + u]);
                float2 z = make_float2(Zpre[rb + u], Zpim[rb + u]);
                #pragma unroll
                for (int d = 0; d < cD; ++d) {
                    float2 g = sG[(k * cT + u) * cD + d];
                    aY[d] = cfma(y, g, aY[d]);
                    aX[d] = cfma(x, g, aX[d]);
                    aZ[d] = cfma(z, g, aZ[d]);
                }
            }
            #pragma unroll
            for (int d = 0; d < cD; ++d) {
                W1[((size_t)n * cT + t) * (cK * cD) + k * cD + d] = aY[d];
                const size_t gi = (((size_t)n * cK + k) * cT + t) * cD + d;
                XpG[gi] = aX[d];
                ZpG[gi] = aZ[d];
            }
        }
    }
}

// =====================================================================
// Kernel 2: tM[n] (64x64 cplx) = sum_k H[n,k]^H @ P[n,k] + beta_u*I  (WMMA)
// block = 512 (16 waves) = 4x4 grid of 16x16 output tiles, grid = N.
// =====================================================================
__global__ __launch_bounds__(512) void k2_tm(
    const float* __restrict__ Hre, const float* __restrict__ Him,
    const float2* __restrict__ P, const float2* __restrict__ beta_u,
    float2* __restrict__ tM, float2* __restrict__ dg)
{
    const int n = blockIdx.x;
    const int wave = threadIdx.x >> 5;
    const int lane = threadIdx.x & 31;
    const int half = lane >> 4;
    const int idx = lane & 15;
    const int m0 = (wave >> 2) * 16;   // output row block (t)
    const int u0 = (wave & 3) * 16;    // output col block (u)

    v8f cre = {0.f, 0.f, 0.f, 0.f, 0.f, 0.f, 0.f, 0.f};
    v8f cim = {0.f, 0.f, 0.f, 0.f, 0.f, 0.f, 0.f, 0.f};

    for (int k = 0; k < cK; ++k) {
        const size_t hb = ((size_t)n * cK + k) * (cR * cT);
        const size_t pb = ((size_t)n * cK + k) * (cR * cT);
        #pragma unroll
        for (int r0 = 0; r0 < cR; r0 += 4) {
            const int ra = r0 + 2 * half;
            // A = H^H tile: A[m][kk] = conj(H[ra+kk][m0+m])
            float hr0 = Hre[hb + (size_t)ra * cT + (m0 + idx)];
            float hi0 = Him[hb + (size_t)ra * cT + (m0 + idx)];
            float hr1 = Hre[hb + (size_t)(ra + 1) * cT + (m0 + idx)];
            float hi1 = Him[hb + (size_t)(ra + 1) * cT + (m0 + idx)];
            v2f ar = {hr0, hr1};
            v2f ai = {-hi0, -hi1};   // imag of H^H
            v2f an = {hi0, hi1};     // -ai  (for Re accumulation)
            // B = P tile: B[kk][j] = P[ra+kk][u0+j]
            float2 p0 = P[pb + (size_t)ra * cT + (u0 + idx)];
            float2 p1 = P[pb + (size_t)(ra + 1) * cT + (u0 + idx)];
            v2f br = {p0.x, p1.x};
            v2f bi = {p0.y, p1.y};
            cre = wmma4(ar, br, cre);   // +Ar*Br
            cre = wmma4(an, bi, cre);   // -Ai*Bi
            cim = wmma4(ar, bi, cim);   // +Ar*Bi
            cim = wmma4(ai, br, cim);   // +Ai*Br
        }
    }

    const float2 bu = beta_u[n];
    #pragma unroll
    for (int v = 0; v < 8; ++v) {
        const int row = m0 + v + 8 * half;
        const int col = u0 + idx;
        float re = cre[v], im = cim[v];
        if (row == col) { re += bu.x; im += bu.y; }
        tM[(size_t)n * cT * cT + (size_t)row * cT + col] = make_float2(re, im);
        if (row == col) dg[(size_t)n * cT + row] = make_float2(re, im);
    }
}

// =====================================================================
// Kernel 3: S = tM @ W1cat (64x64 @ 64x32, WMMA), fuse term1/Zp/Op,
// normalize V[n] and write planar ws + interleaved complex64 output.
// block = 256 (8 waves) = 4x2 grid of 16x16 output tiles, grid = N.
// =====================================================================
__global__ __launch_bounds__(256) void k3_assemble(
    const float2* __restrict__ tM, const float2* __restrict__ W1,
    const float2* __restrict__ XpG, const float2* __restrict__ ZpG,
    const float2* __restrict__ dg,
    const float* __restrict__ Ore, const float* __restrict__ Oim,
    const float* __restrict__ w,
    float* __restrict__ Vre_out, float* __restrict__ Vim_out,
    float* __restrict__ out)
{
    const int n = blockIdx.x;
    const int wave = threadIdx.x >> 5;
    const int lane = threadIdx.x & 31;
    const int half = lane >> 4;
    const int idx = lane & 15;
    const int m0 = (wave >> 1) * 16;   // output row block (t)
    const int j0 = (wave & 1) * 16;    // output col block (k*4+d)

    v8f cre = {0.f, 0.f, 0.f, 0.f, 0.f, 0.f, 0.f, 0.f};
    v8f cim = {0.f, 0.f, 0.f, 0.f, 0.f, 0.f, 0.f, 0.f};

    const size_t tb = (size_t)n * cT * cT;
    const size_t wb = (size_t)n * cT * (cK * cD);
    #pragma unroll 4
    for (int kk0 = 0; kk0 < cT; kk0 += 4) {
        const int c0 = kk0 + 2 * half;
        float2 a0 = tM[tb + (size_t)(m0 + idx) * cT + c0];
        float2 a1 = tM[tb + (size_t)(m0 + idx) * cT + c0 + 1];
        v2f ar = {a0.x, a1.x};
        v2f ai = {a0.y, a1.y};
        v2f an = {-a0.y, -a1.y};
        float2 b0 = W1[wb + (size_t)c0 * (cK * cD) + (j0 + idx)];
        float2 b1 = W1[wb + (size_t)(c0 + 1) * (cK * cD) + (j0 + idx)];
        v2f br = {b0.x, b1.x};
        v2f bi = {b0.y, b1.y};
        cre = wmma4(ar, br, cre);
        cre = wmma4(an, bi, cre);
        cim = wmma4(ar, bi, cim);
        cim = wmma4(ai, br, cim);
    }

    float vre[8], vim[8];
    float pn = 0.f;
    #pragma unroll
    for (int v = 0; v < 8; ++v) {
        const int t = m0 + v + 8 * half;
        const int col = j0 + idx;
        const int k = col >> 2, d = col & 3;
        float2 dgv = dg[(size_t)n * cT + t];
        float inv = 1.0f / (dgv.x * dgv.x + dgv.y * dgv.y);
        float2 idg = make_float2(dgv.x * inv, -dgv.y * inv);  // 1/diag
        const size_t gi = (((size_t)n * cK + k) * cT + t) * cD + d;
        float2 xg = XpG[gi];
        float2 zg = ZpG[gi];
        float2 t1 = cmul(idg, xg);
        float2 el = make_float2(t1.x + cre[v] + zg.x, t1.y + cim[v] + zg.y);
        const float wk = w[k];
        const size_t oi = ((size_t)k * cT + t) * cD + d;
        float re = Ore[oi] + wk * el.x;
        float im = Oim[oi] + wk * el.y;
        vre[v] = re; vim[v] = im;
        pn += re * re + im * im;
    }

    __shared__ float red[256];
    red[threadIdx.x] = pn;
    __syncthreads();
    for (int s = 128; s > 0; s >>= 1) {
        if (threadIdx.x < s) red[threadIdx.x] += red[threadIdx.x + s];
        __syncthreads();
    }
    const float scale = sqrtf(PPOW / red[0]);

    #pragma unroll
    for (int v = 0; v < 8; ++v) {
        const int t = m0 + v + 8 * half;
        const int col = j0 + idx;
        const int k = col >> 2, d = col & 3;
        const size_t vi = (((size_t)n * cK + k) * cT + t) * cD + d;
        float re = vre[v] * scale, im = vim[v] * scale;
        Vre_out[vi] = re;
        Vim_out[vi] = im;
        out[2 * vi] = re;
        out[2 * vi + 1] = im;
    }
}

// =====================================================================
extern "C" void kernel_launch(void* const* d_in, const int* in_sizes, int n_in,
                              void* d_out, int out_size, void* d_ws, size_t ws_size,
                              hipStream_t stream) {
    (void)in_sizes; (void)n_in; (void)out_size; (void)ws_size;
    const float* Hre = (const float*)d_in[0];
    const float* Him = (const float*)d_in[1];
    const float* Vre_in = (const float*)d_in[2];
    const float* Vim_in = (const float*)d_in[3];
    const float* Xre = (const float*)d_in[4];
    const float* Xim = (const float*)d_in[5];
    const float* Yre = (const float*)d_in[6];
    const float* Yim = (const float*)d_in[7];
    const float* Zre = (const float*)d_in[8];
    const float* Zim = (const float*)d_in[9];
    const float* Ore = (const float*)d_in[10];
    const float* Oim = (const float*)d_in[11];
    const float* w   = (const float*)d_in[12];

    // workspace layout (float2 units)
    float2* P   = (float2*)d_ws;                         // N*K*R*T   = 2097152
    float2* G   = P   + (size_t)cN * cK * cR * cT;       // N*K*T*D   = 1048576
    float2* W1  = G   + (size_t)cN * cK * cT * cD;
    float2* XpG = W1  + (size_t)cN * cT * cK * cD;
    float2* ZpG = XpG + (size_t)cN * cK * cT * cD;
    float2* tMw = ZpG + (size_t)cN * cK * cT * cD;       // N*T*T     = 2097152
    float2* dgw = tMw + (size_t)cN * cT * cT;            // N*T
    float2* buw = dgw + (size_t)cN * cT;                 // N
    float*  Vre_ws = (float*)(buw + cN);                 // N*K*T*D floats
    float*  Vim_ws = Vre_ws + (size_t)cN * cK * cT * cD;

    const size_t sliceM = (size_t)cK * cT * cT;  // X/Y/Z per-iteration stride
    const size_t sliceO = (size_t)cK * cT * cD;  // O per-iteration stride

    for (int it = 0; it < cITERS; ++it) {
        const float* vr = (it == 0) ? Vre_in : Vre_ws;
        const float* vi = (it == 0) ? Vim_in : Vim_ws;
        const size_t so = (size_t)it * sliceM;
        const size_t oo = (size_t)it * sliceO;
        k1_small<<<cN, 64, 0, stream>>>(Hre, Him, vr, vi,
                                        Yre + so, Yim + so, Xre + so, Xim + so,
                                        Zre + so, Zim + so, w,
                                        P, G, W1, XpG, ZpG, buw);
        k2_tm<<<cN, 512, 0, stream>>>(Hre, Him, P, buw, tMw, dgw);
        k3_assemble<<<cN, 256, 0, stream>>>(tMw, W1, XpG, ZpG, dgw,
                                            Ore + oo, Oim + oo, w,
                                            Vre_ws, Vim_ws, (float*)d_out);
    }
}